// O2MMatcher_3341484556773
// MI455X (gfx1250) — compile-verified
//
#include <hip/hip_runtime.h>
#include <hip/hip_bf16.h>
#include <float.h>

// Problem constants (from setup_inputs): bs=16, N=4000, C=80, G=300
#define BS     16
#define NP     4000
#define NC     80
#define NG     300
#define TOPK_K 13
#define NITER  16   // ceil(NP/256): per-thread strided elements

// ---- sortable float <-> uint key (monotone order-preserving bijection) ----
__device__ __forceinline__ unsigned f2key(float f) {
  unsigned u = __float_as_uint(f);
  return (u & 0x80000000u) ? ~u : (u | 0x80000000u);
}
__device__ __forceinline__ float key2f(unsigned k) {
  unsigned u = (k & 0x80000000u) ? (k & 0x7fffffffu) : ~k;
  return __uint_as_float(u);
}
__device__ __forceinline__ float sigmoidf(float x) {
  return 1.0f / (1.0f + expf(-x));
}
// x^6 computed the way lax.integer_pow does (x2 * x4)
__device__ __forceinline__ float pow6(float x) {
  float x2 = x * x;
  float x4 = x2 * x2;
  return x2 * x4;
}

__global__ void init_packed_kernel(unsigned long long* __restrict__ packed, int n) {
  int i = blockIdx.x * blockDim.x + threadIdx.x;
  if (i < n) packed[i] = 0ull;
}

// One workgroup per (batch, gt) column. Computes alignment[:,g] into LDS,
// extracts top-13 by iterative argmax (register-resident per-thread maxima,
// only the winner's owner rescans), scatters winning overlaps into a
// per-pred packed-max array via 64-bit global atomicMax.
__global__ __launch_bounds__(256) void col_topk_kernel(
    const float* __restrict__ logits,        // (BS, NP, NC)
    const float* __restrict__ pboxes,        // (BS, NP, 4) cxcywh
    const float* __restrict__ gboxes,        // (BS, NG, 4) cxcywh
    const int*   __restrict__ glabels,       // (BS, NG)
    unsigned long long* __restrict__ packed) // (BS, NP)
{
  const int col = blockIdx.x;
  const int b   = col / NG;
  const int g   = col - b * NG;
  const int tid = threadIdx.x;

  const float*  lg = logits + (size_t)b * NP * NC;
  const float4* pb = (const float4*)pboxes + (size_t)b * NP;

  __shared__ float s_align[NP];   // 16 KB
  __shared__ float s_ov[NP];      // 16 KB
  __shared__ float s_logit[NP];   // 16 KB (filled by async gather)
  __shared__ float rv[8];
  __shared__ int   ri[8];
  __shared__ int   s_win;

  // GT box -> xyxy (same op order as reference)
  float4 gt = ((const float4*)gboxes)[(size_t)b * NG + g];
  const float gx1 = gt.x - 0.5f * gt.z, gy1 = gt.y - 0.5f * gt.w;
  const float gx2 = gt.x + 0.5f * gt.z, gy2 = gt.y + 0.5f * gt.w;
  const float garea = (gx2 - gx1) * (gy2 - gy1);
  const int   label = glabels[(size_t)b * NG + g];

  // Phase 1: kick off async gather of the strided logit column into LDS
  // (CDNA5 ASYNCcnt path), overlapped with the GIoU computation.
  float rov[NITER];
  int cnt = 0;
  for (int n = tid; n < NP; n += 256, ++cnt) {
    unsigned     lds_off = (unsigned)(size_t)&s_logit[n];  // low 32 bits = LDS offset
    const float* gaddr   = lg + (size_t)n * NC + label;
    asm volatile("global_load_async_to_lds_b32 %0, %1, off"
                 :: "v"(lds_off), "v"(gaddr) : "memory");

    if (n + 256 < NP) __builtin_prefetch(&pb[n + 256], 0, 0); // global_prefetch_b8

    float4 p = pb[n];
    float px1 = p.x - 0.5f * p.z, py1 = p.y - 0.5f * p.w;
    float px2 = p.x + 0.5f * p.z, py2 = p.y + 0.5f * p.w;
    float parea = (px2 - px1) * (py2 - py1);
    float ix1 = fmaxf(px1, gx1), iy1 = fmaxf(py1, gy1);
    float ix2 = fminf(px2, gx2), iy2 = fminf(py2, gy2);
    float iw  = fmaxf(ix2 - ix1, 0.0f), ih = fmaxf(iy2 - iy1, 0.0f);
    float inter = iw * ih;
    float uni   = parea + garea - inter;
    float iou   = inter / uni;
    float ex1 = fminf(px1, gx1), ey1 = fminf(py1, gy1);
    float ex2 = fmaxf(px2, gx2), ey2 = fmaxf(py2, gy2);
    float ew  = fmaxf(ex2 - ex1, 0.0f), eh = fmaxf(ey2 - ey1, 0.0f);
    float earea = ew * eh;
    float giou  = iou - (earea - uni) / earea;
    float ov    = -giou;                   // overlaps[n,g]
    s_ov[n]  = ov;
    rov[cnt] = ov;                         // register copy (no LDS re-read)
  }

  // Wait for this wave's async LDS gathers, then build alignment metric and
  // this thread's running (max, argmax) over its strided slice.
  asm volatile("s_wait_asynccnt 0" ::: "memory");
  float bv = -FLT_MAX; int bi = NP;
  cnt = 0;
  for (int n = tid; n < NP; n += 256, ++cnt) {
    float a = sigmoidf(s_logit[n]) * pow6(rov[cnt]);
    s_align[n] = a;
    if (a > bv) { bv = a; bi = n; }        // strict '>' keeps smallest n per thread
  }
  __syncthreads();

  // Phase 2: top-13. Per iteration: wave32 shuffle argmax (no barriers) +
  // 8-partial cross-wave combine; only the winner's owner thread rescans.
  const int lane = tid & 31, wave = tid >> 5;
  for (int k = 0; k < TOPK_K; ++k) {
    float v = bv; int i = bi;
    #pragma unroll
    for (int off = 16; off > 0; off >>= 1) {
      float v2 = __shfl_down(v, off, 32);
      int   i2 = __shfl_down(i, off, 32);
      if (v2 > v || (v2 == v && i2 < i)) { v = v2; i = i2; }
    }
    if (lane == 0) { rv[wave] = v; ri[wave] = i; }
    __syncthreads();
    if (tid == 0) {
      float mv = rv[0]; int mi = ri[0];
      #pragma unroll
      for (int w = 1; w < 8; ++w) {
        float wv = rv[w]; int wi = ri[w];
        if (wv > mv || (wv == mv && wi < mi)) { mv = wv; mi = wi; }
      }
      if (mv > 0.0f && mi < NP) {
        // key: high 32 = sortable overlap, low 32 = 0xFFFFFFFF - g
        // => max picks largest overlap; on ties, smallest g (argmax semantics).
        unsigned long long pk =
            ((unsigned long long)f2key(s_ov[mi]) << 32) |
            (unsigned long long)(0xFFFFFFFFu - (unsigned)g);
        atomicMax(&packed[(size_t)b * NP + mi], pk);
        s_align[mi] = -FLT_MAX;            // remove from further extraction
        s_win = mi;
      } else {
        s_win = -1;                        // all remaining metrics <= 0: stop
      }
    }
    __syncthreads();
    int win = s_win;
    if (win < 0) break;
    if ((win & 255) == tid) {              // only the owner rescans its slice
      bv = -FLT_MAX; bi = NP;
      for (int n = tid; n < NP; n += 256) {
        float a = s_align[n];
        if (a > bv) { bv = a; bi = n; }
      }
    }
  }
}

// Per (b, n): decode winner, emit outputs (ints written as float).
__global__ __launch_bounds__(256) void finalize_kernel(
    const float* __restrict__ logits,
    const int*   __restrict__ glabels,
    const unsigned long long* __restrict__ packed,
    float* __restrict__ out)
{
  int idx = blockIdx.x * blockDim.x + threadIdx.x;   // b*NP + n
  if (idx >= BS * NP) return;
  int b = idx / NP;
  int n = idx - b * NP;

  unsigned long long pk = packed[idx];
  float inds = 0.0f, labf = -1.0f, met = 0.0f;
  if (pk != 0ull) {
    unsigned gwin  = 0xFFFFFFFFu - (unsigned)(pk & 0xFFFFFFFFu);
    float    ov    = key2f((unsigned)(pk >> 32));
    int      label = glabels[(size_t)b * NG + gwin];
    float    sc    = sigmoidf(logits[((size_t)b * NP + n) * NC + label]);
    met  = sc * pow6(ov);                  // alignment[n, argmax]
    inds = (float)(gwin + 1u);
    labf = (float)label;
  }
  const int T = BS * NP;
  out[idx]         = inds;  // assigned_gt_inds
  out[T + idx]     = labf;  // assigned_labels
  out[2 * T + idx] = met;   // assign_metrics
}

extern "C" void kernel_launch(void* const* d_in, const int* in_sizes, int n_in,
                              void* d_out, int out_size, void* d_ws, size_t ws_size,
                              hipStream_t stream) {
  (void)in_sizes; (void)n_in; (void)out_size; (void)ws_size;
  const float* logits  = (const float*)d_in[0];  // pred_logits (16,4000,80) f32
  const float* pboxes  = (const float*)d_in[1];  // pred_boxes  (16,4000,4)  f32
  const float* gboxes  = (const float*)d_in[2];  // gt_boxes    (16,300,4)   f32
  const int*   glabels = (const int*)  d_in[3];  // gt_labels   (16,300)     i32
  float* out = (float*)d_out;

  unsigned long long* packed = (unsigned long long*)d_ws;  // 16*4000*8 = 512 KB
  const int total = BS * NP;

  init_packed_kernel<<<(total + 255) / 256, 256, 0, stream>>>(packed, total);
  col_topk_kernel<<<BS * NG, 256, 0, stream>>>(logits, pboxes, gboxes, glabels, packed);
  finalize_kernel<<<(total + 255) / 256, 256, 0, stream>>>(logits, glabels, packed, out);
}